// BayesianGCNVAE_68865505624427
// MI455X (gfx1250) — compile-verified
//
#include <hip/hip_runtime.h>
#include <math.h>

typedef __attribute__((ext_vector_type(2))) float v2f;
typedef __attribute__((ext_vector_type(8))) float v8f;

__device__ __forceinline__ v8f wmma_f32_16x16x4(v2f a, v2f b, v8f c) {
  // D = A(16x4,f32) * B(4x16,f32) + C(16x16,f32)
  return __builtin_amdgcn_wmma_f32_16x16x4_f32(false, a, false, b, (short)0, c,
                                               false, false);
}

// ---------------------------------------------------------------------------
// One-time weight repack into WMMA lane order:
// out2[((kk*2 + khalf)*DOUT + col)] = { W[kb][col], W[kb+1][col] }, kb = kk*4+khalf*2
// Zero-pads K rows beyond KREAL so the hot loop needs no guards.
// ---------------------------------------------------------------------------
__global__ void repack_b_k(const float* __restrict__ W, float* __restrict__ out2,
                           int KREAL, int DOUT, int NCH) {
  int idx = blockIdx.x * blockDim.x + threadIdx.x;
  const int total = NCH * 2 * DOUT;
  if (idx >= total) return;
  const int kk    = idx / (2 * DOUT);
  const int rem   = idx - kk * 2 * DOUT;
  const int khalf = rem / DOUT;
  const int col   = rem - khalf * DOUT;
  const int kb    = kk * 4 + khalf * 2;
  out2[2 * idx + 0] = (kb     < KREAL) ? W[kb * DOUT + col]       : 0.0f;
  out2[2 * idx + 1] = (kb + 1 < KREAL) ? W[(kb + 1) * DOUT + col] : 0.0f;
}

// ---------------------------------------------------------------------------
// Per-edge MLP: m = relu([x[src], ew] @ Wc + bc); atomic scatter-add to agg[dst]
// One block = 16*ET edges.  DIN/16 waves; each wave owns one 16-col tile and
// iterates ET edge-tiles per B tile (B loaded once -> 4 independent WMMAs).
// ---------------------------------------------------------------------------
template <int DIN, int ET>
__global__ __launch_bounds__((DIN / 16) * 32)
void edge_mlp_wmma(const float* __restrict__ xin, int ldx,
                   const int* __restrict__ src, const int* __restrict__ dst,
                   const float* __restrict__ ew,
                   const v2f* __restrict__ Bp,   // repacked Wc (zero-padded)
                   const float* __restrict__ bc,
                   float* __restrict__ agg, int E) {
  constexpr int KREAL = DIN + 2;
  constexpr int KPAD  = (KREAL + 3) & ~3;
  constexpr int NCH   = KPAD / 4;
  constexpr int NT    = (DIN / 16) * 32;
  constexpr int ROWS  = 16 * ET;
  __shared__ float As[ROWS][KPAD];

  const int tid   = threadIdx.x;
  const int ebase = blockIdx.x * ROWS;

  // Gather ROWS rows of x[src] (contiguous 512B/row) into LDS
  for (int i = tid; i < ROWS * DIN; i += NT) {
    const int er = i / DIN, c = i % DIN;
    const int e  = ebase + er;
    As[er][c] = (e < E) ? xin[(long long)src[e] * ldx + c] : 0.0f;
  }
  for (int er = tid; er < ROWS; er += NT) {
    const int e = ebase + er;
    As[er][DIN]     = (e < E) ? ew[2 * e]     : 0.0f;
    As[er][DIN + 1] = (e < E) ? ew[2 * e + 1] : 0.0f;
    if (KPAD > KREAL) { As[er][KREAL] = 0.0f; As[er][KREAL + 1] = 0.0f; }
  }
  __syncthreads();

  const int wave  = tid >> 5;
  const int lane  = tid & 31;
  const int col0  = wave * 16;
  const int mrow  = lane & 15;   // A row within tile / B col
  const int khalf = lane >> 4;   // K sub-pair selector

  v8f acc[ET];
#pragma unroll
  for (int t = 0; t < ET; ++t) acc[t] = (v8f){};

#pragma unroll
  for (int kk = 0; kk < NCH; ++kk) {
    const int kb = kk * 4 + khalf * 2;
    const v2f b  = Bp[(kk * 2 + khalf) * DIN + col0 + mrow];  // one b64 load
#pragma unroll
    for (int t = 0; t < ET; ++t) {
      v2f a;
      a.x = As[t * 16 + mrow][kb];
      a.y = As[t * 16 + mrow][kb + 1];
      acc[t] = wmma_f32_16x16x4(a, b, acc[t]);  // independent accs: no hazard
    }
  }

  const int   col  = col0 + mrow;
  const float bias = bc[col];
#pragma unroll
  for (int t = 0; t < ET; ++t) {
#pragma unroll
    for (int r = 0; r < 8; ++r) {
      const int e = ebase + t * 16 + r + khalf * 8;  // D row = vgpr + 8*(lane>>4)
      if (e < E) {
        float v = acc[t][r] + bias;
        v = v > 0.0f ? v : 0.0f;  // relu
        atomicAdd(&agg[(long long)dst[e] * DIN + col], v);  // segment_sum
      }
    }
  }
}

// ---------------------------------------------------------------------------
// Per-node GEMM: out = act((agg * inv_deg) @ W + b). ACT: 1 relu, 2 tanh.
// One block = 16*MT nodes; DOUT/16 waves; W tile reused across MT node-tiles.
// ---------------------------------------------------------------------------
template <int DIN, int DOUT, int MT, int ACT>
__global__ __launch_bounds__((DOUT / 16) * 32)
void node_gemm_wmma(const float* __restrict__ agg, const float* __restrict__ inv,
                    const v2f* __restrict__ Wp,   // repacked W
                    const float* __restrict__ bv,
                    float* __restrict__ out, int ldo, int N) {
  constexpr int NCH  = DIN / 4;
  constexpr int NT   = (DOUT / 16) * 32;
  constexpr int ROWS = 16 * MT;
  constexpr int LDA  = DIN + 4;  // pad LDS stride (bank conflicts)
  __shared__ float As[ROWS][LDA];

  const int tid   = threadIdx.x;
  const int nbase = blockIdx.x * ROWS;

  for (int i = tid; i < ROWS * DIN; i += NT) {
    const int nr = i / DIN, c = i % DIN;
    const int n  = nbase + nr;
    As[nr][c] = (n < N) ? agg[(long long)n * DIN + c] * inv[n] : 0.0f;
  }
  __syncthreads();

  const int wave  = tid >> 5;
  const int lane  = tid & 31;
  const int col0  = wave * 16;
  const int mrow  = lane & 15;
  const int khalf = lane >> 4;

  v8f acc[MT];
#pragma unroll
  for (int t = 0; t < MT; ++t) acc[t] = (v8f){};

#pragma unroll
  for (int kk = 0; kk < NCH; ++kk) {
    const int kb = kk * 4 + khalf * 2;
    const v2f b  = Wp[(kk * 2 + khalf) * DOUT + col0 + mrow];
#pragma unroll
    for (int t = 0; t < MT; ++t) {
      v2f a;
      a.x = As[t * 16 + mrow][kb];
      a.y = As[t * 16 + mrow][kb + 1];
      acc[t] = wmma_f32_16x16x4(a, b, acc[t]);
    }
  }

  const int   col  = col0 + mrow;
  const float bias = bv[col];
#pragma unroll
  for (int t = 0; t < MT; ++t) {
#pragma unroll
    for (int r = 0; r < 8; ++r) {
      const int n = nbase + t * 16 + r + khalf * 8;
      if (n < N) {
        float v = acc[t][r] + bias;
        if (ACT == 1)      v = v > 0.0f ? v : 0.0f;
        else if (ACT == 2) v = tanhf(v);
        out[(long long)n * ldo + col] = v;
      }
    }
  }
}

// ---------------------------------------------------------------------------
// Helpers
// ---------------------------------------------------------------------------
__global__ void zero_f32(float* __restrict__ p, long long n) {
  long long i = (long long)blockIdx.x * blockDim.x + threadIdx.x;
  if (i < n) p[i] = 0.0f;
}

__global__ void degree_k(const int* __restrict__ dst, float* __restrict__ cnt, int E) {
  int e = blockIdx.x * blockDim.x + threadIdx.x;
  if (e < E) atomicAdd(&cnt[dst[e]], 1.0f);
}

__global__ void recip_k(float* __restrict__ cnt, int N) {
  int n = blockIdx.x * blockDim.x + threadIdx.x;
  if (n < N) cnt[n] = 1.0f / fmaxf(cnt[n], 1.0f);
}

__global__ void split_k(const float* __restrict__ h2, float* __restrict__ mu,
                        float* __restrict__ lv, int N) {
  long long i = (long long)blockIdx.x * blockDim.x + threadIdx.x;
  if (i < (long long)N * 64) {
    const long long n = i >> 6, c = i & 63;
    mu[i] = h2[n * 128 + c];
    lv[i] = h2[n * 128 + 64 + c];
  }
}

__global__ void pet_k(const float* __restrict__ x, const int* __restrict__ src,
                      const int* __restrict__ dst, const float* __restrict__ ew,
                      const float* __restrict__ etW, const float* __restrict__ etb,
                      float* __restrict__ pet, int E) {
  int e = blockIdx.x * blockDim.x + threadIdx.x;
  if (e >= E) return;
  const float4* xs = (const float4*)(x + (long long)src[e] * 128);
  const float4* xd = (const float4*)(x + (long long)dst[e] * 128);
  const float4* w4 = (const float4*)etW;
  float acc = etb[0];
#pragma unroll 4
  for (int c = 0; c < 32; ++c) {
    float4 a = xs[c], b = xd[c], w = w4[c];
    acc += fabsf(a.x - b.x) * w.x + fabsf(a.y - b.y) * w.y +
           fabsf(a.z - b.z) * w.z + fabsf(a.w - b.w) * w.w;
  }
  acc += ew[2 * e] * etW[128] + ew[2 * e + 1] * etW[129];
  pet[e] = acc;
}

// ---------------------------------------------------------------------------
extern "C" void kernel_launch(void* const* d_in, const int* in_sizes, int n_in,
                              void* d_out, int out_size, void* d_ws, size_t ws_size,
                              hipStream_t stream) {
  const float* x     = (const float*)d_in[0];
  const int*   ei    = (const int*)d_in[1];
  const float* ew    = (const float*)d_in[2];
  const float* c1_Wc = (const float*)d_in[3];
  const float* c1_bc = (const float*)d_in[4];
  const float* c1_W  = (const float*)d_in[5];
  const float* c1_b  = (const float*)d_in[6];
  const float* c2_Wc = (const float*)d_in[7];
  const float* c2_bc = (const float*)d_in[8];
  const float* c2_W  = (const float*)d_in[9];
  const float* c2_b  = (const float*)d_in[10];
  const float* c3_Wc = (const float*)d_in[11];
  const float* c3_bc = (const float*)d_in[12];
  const float* c3_W  = (const float*)d_in[13];
  const float* c3_b  = (const float*)d_in[14];
  const float* c4_Wc = (const float*)d_in[15];
  const float* c4_bc = (const float*)d_in[16];
  const float* c4_W  = (const float*)d_in[17];
  const float* c4_b  = (const float*)d_in[18];
  const float* etW   = (const float*)d_in[19];
  const float* etb   = (const float*)d_in[20];

  const int F = 128, L = 64;
  const int N = in_sizes[0] / F;
  const int E = in_sizes[1] / 2;
  const int* src = ei;
  const int* dst = ei + E;

  // pack-buffer chunk counts
  const int NCH130 = 33;  // (130 padded to 132)/4  -> edge Wc, din=128
  const int NCH66  = 17;  // (66 padded to 68)/4    -> edge Wc, din=64
  const int NCH128 = 32;  // node W, din=128
  const int NCH64  = 16;  // node W, din=64

  // workspace layout (floats)
  float* w    = (float*)d_ws;
  float* inv  = w;                               // N
  float* agg  = inv + N;                         // N*128
  float* h1   = agg + (long long)N * 128;        // N*128
  float* h2   = h1 + (long long)N * 128;         // N*128
  float* d3   = h1;                              // h1 dead after conv2 edge pass
  float* pk   = h2 + (long long)N * 128;
  float* p1wc = pk;              pk += NCH130 * 2 * 128 * 2;
  float* p2wc = pk;              pk += NCH130 * 2 * 128 * 2;
  float* p3wc = pk;              pk += NCH66  * 2 * 64  * 2;
  float* p4wc = pk;              pk += NCH130 * 2 * 128 * 2;
  float* p1w  = pk;              pk += NCH128 * 2 * 128 * 2;
  float* p2w  = pk;              pk += NCH128 * 2 * 128 * 2;
  float* p3w  = pk;              pk += NCH64  * 2 * 128 * 2;
  float* p4w  = pk;              pk += NCH128 * 2 * 128 * 2;

  float* recon = (float*)d_out;                 // N*F
  float* mu    = recon + (long long)N * F;      // N*L
  float* lv    = mu + (long long)N * L;         // N*L
  float* pet   = lv + (long long)N * L;         // E

  const int T = 256;
  auto cdiv = [](long long a, long long b) { return (int)((a + b - 1) / b); };
  constexpr int ET = 4;  // edge-tiles per block
  constexpr int MT = 4;  // node-tiles per block
  const int eg = cdiv(E, 16 * ET);
  const int ng = cdiv(N, 16 * MT);

  // --- repack all weights into WMMA lane order (cheap, one pass) ---
  auto rp = [&](const float* Wsrc, float* dst2, int KREAL, int DOUT, int NCHx) {
    const int tot = NCHx * 2 * DOUT;
    repack_b_k<<<cdiv(tot, T), T, 0, stream>>>(Wsrc, dst2, KREAL, DOUT, NCHx);
  };
  rp(c1_Wc, p1wc, 130, 128, NCH130);
  rp(c2_Wc, p2wc, 130, 128, NCH130);
  rp(c3_Wc, p3wc, 66,  64,  NCH66);
  rp(c4_Wc, p4wc, 130, 128, NCH130);
  rp(c1_W,  p1w,  128, 128, NCH128);
  rp(c2_W,  p2w,  128, 128, NCH128);
  rp(c3_W,  p3w,  64,  128, NCH64);
  rp(c4_W,  p4w,  128, 128, NCH128);

  // degrees (layer-invariant) -> inv = 1/max(cnt,1)
  zero_f32<<<cdiv(N, T), T, 0, stream>>>(inv, N);
  degree_k<<<cdiv(E, T), T, 0, stream>>>(dst, inv, E);
  recip_k<<<cdiv(N, T), T, 0, stream>>>(inv, N);

  // conv1: x -> h1 (relu)
  zero_f32<<<cdiv((long long)N * 128, T), T, 0, stream>>>(agg, (long long)N * 128);
  edge_mlp_wmma<128, ET><<<eg, 256, 0, stream>>>(x, 128, src, dst, ew, (const v2f*)p1wc, c1_bc, agg, E);
  node_gemm_wmma<128, 128, MT, 1><<<ng, 256, 0, stream>>>(agg, inv, (const v2f*)p1w, c1_b, h1, 128, N);

  // conv2: h1 -> h2 (relu), then split mu/logvar
  zero_f32<<<cdiv((long long)N * 128, T), T, 0, stream>>>(agg, (long long)N * 128);
  edge_mlp_wmma<128, ET><<<eg, 256, 0, stream>>>(h1, 128, src, dst, ew, (const v2f*)p2wc, c2_bc, agg, E);
  node_gemm_wmma<128, 128, MT, 1><<<ng, 256, 0, stream>>>(agg, inv, (const v2f*)p2w, c2_b, h2, 128, N);
  split_k<<<cdiv((long long)N * 64, T), T, 0, stream>>>(h2, mu, lv, N);

  // conv3: z = mu = h2[:, :64] (ld=128) -> d3 (relu)
  zero_f32<<<cdiv((long long)N * 64, T), T, 0, stream>>>(agg, (long long)N * 64);
  edge_mlp_wmma<64, ET><<<eg, 128, 0, stream>>>(h2, 128, src, dst, ew, (const v2f*)p3wc, c3_bc, agg, E);
  node_gemm_wmma<64, 128, MT, 1><<<ng, 256, 0, stream>>>(agg, inv, (const v2f*)p3w, c3_b, d3, 128, N);

  // conv4: d3 -> recon (tanh)
  zero_f32<<<cdiv((long long)N * 128, T), T, 0, stream>>>(agg, (long long)N * 128);
  edge_mlp_wmma<128, ET><<<eg, 256, 0, stream>>>(d3, 128, src, dst, ew, (const v2f*)p4wc, c4_bc, agg, E);
  node_gemm_wmma<128, 128, MT, 2><<<ng, 256, 0, stream>>>(agg, inv, (const v2f*)p4w, c4_b, recon, 128, N);

  // edge-score head
  pet_k<<<cdiv(E, T), T, 0, stream>>>(x, src, dst, ew, etW, etb, pet, E);
}